// GCN_Capacity_20289425507112
// MI455X (gfx1250) — compile-verified
//
#include <hip/hip_runtime.h>
#include <math.h>

typedef __attribute__((ext_vector_type(2))) float v2f;
typedef __attribute__((ext_vector_type(8))) float v8f;

#define NN 50000
#define NE 800000
#define COUT 64

// ---------------- degree / normalization ----------------
__global__ void k_init_deg(float* __restrict__ deg) {
    int i = blockIdx.x * blockDim.x + threadIdx.x;
    if (i < NN) deg[i] = 1.0f;  // self-loop
}

__global__ void k_count_deg(const long long* __restrict__ dst, float* __restrict__ deg) {
    int e = blockIdx.x * blockDim.x + threadIdx.x;
    if (e < NE) atomicAdd(&deg[(int)dst[e]], 1.0f);
}

__global__ void k_dinv(const float* __restrict__ deg, float* __restrict__ dinv) {
    int i = blockIdx.x * blockDim.x + threadIdx.x;
    if (i < NN) dinv[i] = rsqrtf(deg[i]);
}

// ---------------- WMMA fp32 GEMM: H[N,64] = X[N,Cin] @ W[Cin,64] ----------------
// One wave computes a 16x64 output tile as four 16x16 f32 accumulators,
// stepping K by 4 with V_WMMA_F32_16X16X4_F32.
__global__ __launch_bounds__(256) void k_gemm_wmma(const float* __restrict__ X,
                                                   const float* __restrict__ W,
                                                   float* __restrict__ H, int Cin) {
    const int wave = threadIdx.x >> 5;
    const int lane = threadIdx.x & 31;
    const int tile = blockIdx.x * 8 + wave;          // 16-row tile index
    if (tile * 16 >= NN) return;                     // wave-uniform: EXEC stays all-1s
    const int row0 = tile * 16;

    const int m  = lane & 15;            // A: M row, B: N col (within 16-tile)
    const int kb = (lane >> 4) << 1;     // lane half selects K pair {0,1} or {2,3}

    v8f acc0 = {}, acc1 = {}, acc2 = {}, acc3 = {};
    const float* xrow = X + (size_t)(row0 + m) * Cin;

    for (int k0 = 0; k0 < Cin; k0 += 4) {
        // A 16x4: lane m holds (k0+kb, k0+kb+1)
        v2f a;
        a.x = xrow[k0 + kb];
        a.y = xrow[k0 + kb + 1];
        const float* w0 = W + (size_t)(k0 + kb) * COUT;
        const float* w1 = W + (size_t)(k0 + kb + 1) * COUT;
        // B 4x16 per 16-col tile: lane n holds rows (k0+kb, k0+kb+1)
        v2f b0, b1, b2, b3;
        b0.x = w0[m];      b0.y = w1[m];
        b1.x = w0[m + 16]; b1.y = w1[m + 16];
        b2.x = w0[m + 32]; b2.y = w1[m + 32];
        b3.x = w0[m + 48]; b3.y = w1[m + 48];
        acc0 = __builtin_amdgcn_wmma_f32_16x16x4_f32(false, a, false, b0, (short)0, acc0, false, false);
        acc1 = __builtin_amdgcn_wmma_f32_16x16x4_f32(false, a, false, b1, (short)0, acc1, false, false);
        acc2 = __builtin_amdgcn_wmma_f32_16x16x4_f32(false, a, false, b2, (short)0, acc2, false, false);
        acc3 = __builtin_amdgcn_wmma_f32_16x16x4_f32(false, a, false, b3, (short)0, acc3, false, false);
    }

    // C/D layout: VGPR r -> lanes 0-15: M=r, lanes 16-31: M=r+8 ; lane%16 = N
    float* hrow = H + (size_t)row0 * COUT;
    const int nl = lane & 15;
    const int mh = (lane >> 4) * 8;
    #pragma unroll
    for (int r = 0; r < 8; ++r) {
        const size_t mm = (size_t)(mh + r) * COUT;
        hrow[mm + nl]      = acc0[r];
        hrow[mm + nl + 16] = acc1[r];
        hrow[mm + nl + 32] = acc2[r];
        hrow[mm + nl + 48] = acc3[r];
    }
}

// ---------------- aggregation ----------------
// agg[i] = h[i] * dinv[i]^2   (self-loop contribution; also zero-initializes agg)
__global__ void k_self(const float* __restrict__ h, const float* __restrict__ dinv,
                       float* __restrict__ agg) {
    int i = blockIdx.x * blockDim.x + threadIdx.x;
    if (i < NN * COUT) {
        float di = dinv[i >> 6];
        agg[i] = h[i] * di * di;
    }
}

// edge scatter: blockDim = (64, 4); lanes cover 64 channels of one edge
__global__ void k_scatter(const long long* __restrict__ src, const long long* __restrict__ dst,
                          const float* __restrict__ dinv, const float* __restrict__ h,
                          float* __restrict__ agg) {
    int e = blockIdx.x * 4 + threadIdx.y;
    if (e >= NE) return;
    int s = (int)src[e];
    int d = (int)dst[e];
    float nrm = dinv[s] * dinv[d];
    int c = threadIdx.x;
    atomicAdd(&agg[(size_t)d * COUT + c], h[(size_t)s * COUT + c] * nrm);
}

// out[i] = act(agg[i] + b[c]) ; exact GELU via erf for hidden layers
__global__ void k_bias_act(const float* __restrict__ agg, const float* __restrict__ b,
                           float* __restrict__ out, int do_gelu) {
    int i = blockIdx.x * blockDim.x + threadIdx.x;
    if (i < NN * COUT) {
        float v = agg[i] + b[i & 63];
        if (do_gelu) v = 0.5f * v * (1.0f + erff(v * 0.70710678118654752f));
        out[i] = v;
    }
}

extern "C" void kernel_launch(void* const* d_in, const int* in_sizes, int n_in,
                              void* d_out, int out_size, void* d_ws, size_t ws_size,
                              hipStream_t stream) {
    const float*     x    = (const float*)d_in[0];
    const long long* ei   = (const long long*)d_in[1];   // int64 [2, NE]
    const float*     W1   = (const float*)d_in[2];
    const float*     b1   = (const float*)d_in[3];
    const float*     W2   = (const float*)d_in[4];
    const float*     b2   = (const float*)d_in[5];
    const float*     W3   = (const float*)d_in[6];
    const float*     b3   = (const float*)d_in[7];
    float* out = (float*)d_out;

    const long long* src = ei;
    const long long* dst = ei + NE;

    // workspace layout
    float* deg  = (float*)d_ws;            // NN
    float* dinv = deg + NN;                // NN
    float* h    = dinv + NN;               // NN*64
    float* agg  = h + (size_t)NN * COUT;   // NN*64 (activations written in-place)

    const int TPB = 256;
    const int gN   = (NN + TPB - 1) / TPB;
    const int gE   = (NE + TPB - 1) / TPB;
    const int gNC  = (NN * COUT + TPB - 1) / TPB;
    const int gGemm = ((NN / 16) + 7) / 8;        // 8 waves (tiles) per block
    dim3 sBlock(64, 4);
    const int gScat = (NE + 3) / 4;

    // normalization (once)
    k_init_deg<<<gN, TPB, 0, stream>>>(deg);
    k_count_deg<<<gE, TPB, 0, stream>>>(dst, deg);
    k_dinv<<<gN, TPB, 0, stream>>>(deg, dinv);

    // layer 1: Cin=128, GELU
    k_gemm_wmma<<<gGemm, TPB, 0, stream>>>(x, W1, h, 128);
    k_self<<<gNC, TPB, 0, stream>>>(h, dinv, agg);
    k_scatter<<<gScat, sBlock, 0, stream>>>(src, dst, dinv, h, agg);
    k_bias_act<<<gNC, TPB, 0, stream>>>(agg, b1, agg, 1);

    // layer 2: Cin=64, GELU
    k_gemm_wmma<<<gGemm, TPB, 0, stream>>>(agg, W2, h, 64);
    k_self<<<gNC, TPB, 0, stream>>>(h, dinv, agg);
    k_scatter<<<gScat, sBlock, 0, stream>>>(src, dst, dinv, h, agg);
    k_bias_act<<<gNC, TPB, 0, stream>>>(agg, b2, agg, 1);

    // layer 3: Cin=64, no activation, write d_out
    k_gemm_wmma<<<gGemm, TPB, 0, stream>>>(agg, W3, h, 64);
    k_self<<<gNC, TPB, 0, stream>>>(h, dinv, out);
    k_scatter<<<gScat, sBlock, 0, stream>>>(src, dst, dinv, h, out);
    k_bias_act<<<gNC, TPB, 0, stream>>>(out, b3, out, 0);
}